// SFPM_63651415327293
// MI455X (gfx1250) — compile-verified
//
#include <hip/hip_runtime.h>
#include <hip/hip_bf16.h>

// ---------------------------------------------------------------------------
// MI455X (gfx1250) sparse focal modulation block.
// Memory-bound (~2.6 GB traffic vs ~23 GFLOP at 23.3 TB/s -> ~110 us), so:
//  * fp32 throughout, V_WMMA_F32_16X16X4_F32 for all 64x64 channel GEMMs
//  * all affine-of-x branches folded (Q, G, imp-hidden, x_proj)
//  * focal gating applied in the final fused kernel (no K_agg buffer)
//  * BN stats fused into the WMMA GEMM via deterministic shuffle trees
//  * weight tiles staged to LDS with GLOBAL_LOAD_ASYNC_TO_LDS_B128 (ASYNCcnt)
//  * no floating-point atomics anywhere (bitwise deterministic)
// ---------------------------------------------------------------------------

#define B_    8
#define H_    224
#define W_    224
#define HW    50176            // 224*224
#define NPIX  401408           // B*HW
#define NTOT  25690112u        // NPIX*64
#define KSEL  15052            // max(int(0.3*HW),64)
#define TILEW 25088            // NPIX/16 : 16-pixel wave tiles

typedef float v2f __attribute__((ext_vector_type(2)));
typedef float v8f __attribute__((ext_vector_type(8)));

// ---------------------------------------------------------------------------
// async global->LDS staging (CDNA5 GLOBAL_LOAD_ASYNC_TO_LDS_B128, ASYNCcnt)
// ---------------------------------------------------------------------------
__device__ __forceinline__ unsigned lds_off_u32(const void* p) {
    return (unsigned)(unsigned long long)(__attribute__((address_space(3))) const void*)p;
}
__device__ __forceinline__ void async_copy16(float* lds_dst, const float* gsrc) {
#if defined(__HIP_DEVICE_COMPILE__) && defined(__gfx1250__)
    unsigned lo = lds_off_u32(lds_dst);
    unsigned long long ga = (unsigned long long)(const void*)gsrc;
    asm volatile("global_load_async_to_lds_b128 %0, %1, off" :: "v"(lo), "v"(ga) : "memory");
#else
    lds_dst[0] = gsrc[0]; lds_dst[1] = gsrc[1];
    lds_dst[2] = gsrc[2]; lds_dst[3] = gsrc[3];
#endif
}
__device__ __forceinline__ void async_wait0() {
#if defined(__HIP_DEVICE_COMPILE__) && defined(__gfx1250__)
    asm volatile("s_wait_asynccnt 0x0" ::: "memory");
#endif
}

// ---------------------------------------------------------------------------
// small helpers
// ---------------------------------------------------------------------------
__global__ void k_zero_u32(unsigned* p, int n) {
    int i = blockIdx.x * blockDim.x + threadIdx.x;
    if (i < n) p[i] = 0u;
}

// fold: imp hidden affines, fin_w affine of x
__global__ void k_prep(const float* w_in, const float* b_in,
                       const float* imp_w1, const float* imp_b1,
                       const float* fin_w, const float* fin_b,
                       float* alpha, float* beta, float* Afin, float* Bfin) {
    int t = threadIdx.x;
    if (t < 16) {
        float a = 0.f, b = 0.f;
        for (int c = 0; c < 64; ++c) {
            float w = imp_w1[t * 64 + c];
            a += w * w_in[64 + c];
            b += w * b_in[64 + c];
        }
        alpha[t] = a;
        beta[t]  = b + imp_b1[t];
    }
    if (t < 64) {
        float a = 0.f, b = 0.f;
        for (int c = 0; c < 64; ++c) {
            float w = fin_w[t * 64 + c];
            a += w * w_in[c];
            b += w * b_in[c];
        }
        Afin[t] = a;
        Bfin[t] = b + fin_b[t];
    }
}

// ---------------------------------------------------------------------------
// importance map: sigmoid(3x3 conv over 16 folded relu affines of x)
// ---------------------------------------------------------------------------
__global__ void k_imp(const float* __restrict__ x,
                      const float* __restrict__ alpha,
                      const float* __restrict__ beta,
                      const float* __restrict__ iw2,
                      const float* __restrict__ ib2,
                      float* __restrict__ imp) {
    int p = blockIdx.x * blockDim.x + threadIdx.x;
    if (p >= NPIX) return;
    int b = p / HW, hw = p - b * HW;
    int h = hw / W_, w = hw - h * W_;
    float acc = ib2[0];
    for (int dy = -1; dy <= 1; ++dy) {
        int hh = h + dy;
        if (hh < 0 || hh >= H_) continue;
        for (int dx = -1; dx <= 1; ++dx) {
            int ww = w + dx;
            if (ww < 0 || ww >= W_) continue;
            float xn  = x[(size_t)b * HW + hh * W_ + ww];
            int   off = (dy + 1) * 3 + (dx + 1);
            for (int j = 0; j < 16; ++j) {
                float hv = xn * alpha[j] + beta[j];
                hv = hv > 0.f ? hv : 0.f;
                acc += iw2[j * 9 + off] * hv;
            }
        }
    }
    imp[p] = 1.f / (1.f + __expf(-acc));
}

// ---------------------------------------------------------------------------
// exact top-k threshold: two-level 16-bit radix select (positive floats)
// ---------------------------------------------------------------------------
__global__ void k_hist_hi(const float* imp, unsigned* hist) {
    int i = blockIdx.x * blockDim.x + threadIdx.x;
    if (i >= NPIX) return;
    int b = i / HW;
    unsigned u = __float_as_uint(imp[i]);
    atomicAdd(&hist[(size_t)b * 65536 + (u >> 16)], 1u);
}
__global__ void k_pick_hi(const unsigned* hist, unsigned* picks) {
    int b = blockIdx.x;
    if (threadIdx.x != 0) return;
    const unsigned* h = hist + (size_t)b * 65536;
    unsigned cum = 0, bucket = 0, gt = 0;
    for (int i = 65535; i >= 0; --i) {
        unsigned c = h[i];
        if (cum + c >= (unsigned)KSEL) { bucket = (unsigned)i; gt = cum; break; }
        cum += c;
    }
    picks[b * 4 + 0] = bucket;
    picks[b * 4 + 1] = gt;
}
__global__ void k_hist_lo(const float* imp, const unsigned* picks, unsigned* hist) {
    int i = blockIdx.x * blockDim.x + threadIdx.x;
    if (i >= NPIX) return;
    int b = i / HW;
    unsigned u = __float_as_uint(imp[i]);
    if ((u >> 16) == picks[b * 4 + 0])
        atomicAdd(&hist[(size_t)b * 65536 + (u & 0xFFFFu)], 1u);
}
__global__ void k_pick_lo(const unsigned* hist, unsigned* picks) {
    int b = blockIdx.x;
    if (threadIdx.x != 0) return;
    const unsigned* h = hist + (size_t)b * 65536;
    unsigned gtHi = picks[b * 4 + 1], bucket = picks[b * 4 + 0];
    unsigned cum = 0;
    for (int j = 65535; j >= 0; --j) {
        unsigned c = h[j];
        if (gtHi + cum + c >= (unsigned)KSEL) {
            picks[b * 4 + 2] = (bucket << 16) | (unsigned)j;   // threshold bits
            picks[b * 4 + 3] = (unsigned)KSEL - gtHi - cum;    // ties to accept
            return;
        }
        cum += c;
    }
    picks[b * 4 + 2] = 0u;
    picks[b * 4 + 3] = 0u;
}
// deterministic index-ordered tie-break: one block per batch, block scan
__global__ void k_mask(const float* imp, const unsigned* picks, float* mask) {
    int b = blockIdx.x;
    unsigned T = picks[b * 4 + 2], needEq = picks[b * 4 + 3];
    __shared__ unsigned scan[1024];
    __shared__ unsigned running;
    if (threadIdx.x == 0) running = 0;
    __syncthreads();
    for (int base = 0; base < HW; base += 1024) {   // HW = 49*1024 exactly
        int i = base + threadIdx.x;
        unsigned u  = __float_as_uint(imp[(size_t)b * HW + i]);
        unsigned eq = (u == T) ? 1u : 0u;
        scan[threadIdx.x] = eq;
        __syncthreads();
        for (int s = 1; s < 1024; s <<= 1) {
            unsigned v = (threadIdx.x >= (unsigned)s) ? scan[threadIdx.x - s] : 0u;
            __syncthreads();
            scan[threadIdx.x] += v;
            __syncthreads();
        }
        unsigned incl = scan[threadIdx.x];
        unsigned excl = incl - eq;
        unsigned r    = running;
        float m = (u > T) ? 1.f : ((eq && (r + excl < needEq)) ? 1.f : 0.f);
        mask[(size_t)b * HW + i] = m;
        __syncthreads();
        if (threadIdx.x == 1023) running = r + incl;
        __syncthreads();
    }
}

// K planes = masked affines of x  (K only ever used masked)
__global__ void k_writeKmasked(const float* __restrict__ x,
                               const float* __restrict__ mask,
                               const float* __restrict__ w_in,
                               const float* __restrict__ b_in,
                               float* __restrict__ Kbuf) {
    int p = blockIdx.x * blockDim.x + threadIdx.x;
    if (p >= NPIX) return;
    float xv = x[p], mv = mask[p];
    int b = p / HW, hw = p - b * HW;
    size_t base = (size_t)b * 64 * HW + hw;
    for (int c = 0; c < 64; ++c)
        Kbuf[base + (size_t)c * HW] = (xv * w_in[64 + c] + b_in[64 + c]) * mv;
}

// ---------------------------------------------------------------------------
// depthwise conv
// ---------------------------------------------------------------------------
__global__ void k_dwconv(const float* __restrict__ src,
                         float* __restrict__ dst,
                         const float* __restrict__ dw, int ksz, int dil) {
    size_t i = (size_t)blockIdx.x * blockDim.x + threadIdx.x;
    if (i >= (size_t)NTOT) return;
    int hw = (int)(i % HW);
    size_t t = i / HW;
    int c = (int)(t % 64), b = (int)(t / 64);
    int h = hw / W_, w = hw - h * W_;
    int half = ksz / 2;
    const float* wt = dw + c * ksz * ksz;
    size_t plane = ((size_t)b * 64 + c) * HW;
    float acc = 0.f;
    for (int ky = 0; ky < ksz; ++ky) {
        int hh = h + (ky - half) * dil;
        if (hh < 0 || hh >= H_) continue;
        for (int kx = 0; kx < ksz; ++kx) {
            int ww = w + (kx - half) * dil;
            if (ww < 0 || ww >= W_) continue;
            acc += wt[ky * ksz + kx] * src[plane + hh * W_ + ww];
        }
    }
    dst[i] = acc;
}

// ---------------------------------------------------------------------------
// pointwise 64x64 GEMM over pixels via V_WMMA_F32_16X16X4_F32, with fused
// deterministic per-wave BN partial sums (shuffle trees, fixed order).
// Fragmenting (ISA 7.12.2): A(16x4): lane&15=M, hi half K+2;
// B(4x16): lane&15=N(pixel), hi half K+2; D: VGPR v -> M=v / v+8, N=lane&15.
// ---------------------------------------------------------------------------
__global__ void __launch_bounds__(256) k_pw_wmma(const float* __restrict__ src,
                                                 float* __restrict__ dst,
                                                 const float* __restrict__ Wgt,
                                                 const float* __restrict__ bias,
                                                 float* __restrict__ partS,
                                                 float* __restrict__ partQ) {
    __shared__ float sW[64 * 64];
    __shared__ float sB[64];
    int tid = threadIdx.x;
    for (int i = tid * 4; i < 4096; i += 256 * 4)
        async_copy16(&sW[i], &Wgt[i]);                 // ASYNCcnt-tracked
    if (tid < 64) sB[tid] = bias[tid];
    async_wait0();
    __syncthreads();

    int wave = tid >> 5, lane = tid & 31, hi = lane >> 4, ln = lane & 15;
    int wt   = blockIdx.x * 8 + wave;                  // wave tile id
    size_t p0 = (size_t)wt * 16;                       // NPIX/16 tiles exactly
    int b   = (int)(p0 / HW);
    int hw0 = (int)(p0 - (size_t)b * HW);
    const float* sp = src + (size_t)b * 64 * HW + hw0 + ln;
    float*       dp = dst + (size_t)b * 64 * HW + hw0 + ln;

    v8f acc[4];
#pragma unroll
    for (int m = 0; m < 4; ++m)
#pragma unroll
        for (int v = 0; v < 8; ++v) acc[m][v] = sB[m * 16 + v + 8 * hi];

    for (int k0 = 0; k0 < 64; k0 += 4) {
        v2f bf;
        bf.x = sp[(size_t)(k0 + 2 * hi) * HW];
        bf.y = sp[(size_t)(k0 + 1 + 2 * hi) * HW];
#pragma unroll
        for (int m = 0; m < 4; ++m) {
            v2f af;
            af.x = sW[(m * 16 + ln) * 64 + k0 + 2 * hi];
            af.y = sW[(m * 16 + ln) * 64 + k0 + 2 * hi + 1];
            acc[m] = __builtin_amdgcn_wmma_f32_16x16x4_f32(
                false, af, false, bf, (short)0, acc[m], false, false);
        }
    }
#pragma unroll
    for (int m = 0; m < 4; ++m)
#pragma unroll
        for (int v = 0; v < 8; ++v) {
            float val = acc[m][v];
            dp[(size_t)(m * 16 + v + 8 * hi) * HW] = val;
            // deterministic 16-lane shuffle-tree over the pixel dimension
            float vs = val, vq = val * val;
#pragma unroll
            for (int d = 1; d < 16; d <<= 1) {
                vs += __shfl_xor(vs, d, 32);
                vq += __shfl_xor(vq, d, 32);
            }
            if (ln == 0) {
                int ch = m * 16 + v + 8 * hi;
                partS[(size_t)wt * 64 + ch] = vs;
                partQ[(size_t)wt * 64 + ch] = vq;
            }
        }
}

// fixed-order tree reduction of per-wave partials -> per-channel sum/sumsq
__global__ void k_stats_red(const float* __restrict__ partS,
                            const float* __restrict__ partQ,
                            float* sum, float* sumsq) {
    int c = blockIdx.x;
    __shared__ float s1[256], s2[256];
    float a = 0.f, q = 0.f;
    for (int t = threadIdx.x; t < TILEW; t += 256) {
        a += partS[(size_t)t * 64 + c];
        q += partQ[(size_t)t * 64 + c];
    }
    s1[threadIdx.x] = a;
    s2[threadIdx.x] = q;
    __syncthreads();
    for (int s = 128; s > 0; s >>= 1) {
        if (threadIdx.x < s) {
            s1[threadIdx.x] += s1[threadIdx.x + s];
            s2[threadIdx.x] += s2[threadIdx.x + s];
        }
        __syncthreads();
    }
    if (threadIdx.x == 0) { sum[c] = s1[0]; sumsq[c] = s2[0]; }
}
__global__ void k_bnparam(const float* sum, const float* sumsq,
                          const float* g, const float* bb,
                          float* scale, float* shift) {
    int c = threadIdx.x;
    if (c >= 64) return;
    const float N = (float)NPIX;
    float mu  = sum[c] / N;
    float var = sumsq[c] / N - mu * mu;
    float sc  = g[c] * rsqrtf(var + 1e-5f);
    scale[c] = sc;
    shift[c] = bb[c] - mu * sc;
}
// elementwise bn+relu -> focal level output
__global__ void k_bnrelu(const float* __restrict__ y,
                         const float* __restrict__ scale,
                         const float* __restrict__ shift,
                         float* __restrict__ fcur) {
    size_t i = (size_t)blockIdx.x * blockDim.x + threadIdx.x;
    if (i >= (size_t)NTOT) return;
    int c = (int)((i / HW) % 64);
    float f = y[i] * scale[c] + shift[c];
    fcur[i] = f > 0.f ? f : 0.f;
}
// per-(b,c) plane sums of focal2 (fixed-order tree, no atomics)
__global__ void k_kgsum(const float* __restrict__ f2, float* __restrict__ Kg) {
    int bc = blockIdx.x;                       // b*64+c
    __shared__ float s1[256];
    const float* pl = f2 + (size_t)bc * HW;
    float a = 0.f;
    for (int t = threadIdx.x; t < HW; t += 256) a += pl[t];
    s1[threadIdx.x] = a;
    __syncthreads();
    for (int s = 128; s > 0; s >>= 1) {
        if (threadIdx.x < s) s1[threadIdx.x] += s1[threadIdx.x + s];
        __syncthreads();
    }
    if (threadIdx.x == 0) Kg[bc] = s1[0];
}

// ---------------------------------------------------------------------------
// final fused kernel: gate focal0..2 + global -> K_final -> V (WMMA) -> QV ->
// O (WMMA) -> + folded x_proj.  QV re-fragmented D->B through per-wave LDS.
// ---------------------------------------------------------------------------
__global__ void __launch_bounds__(256) k_final(const float* __restrict__ f0,
                                               const float* __restrict__ f1,
                                               const float* __restrict__ f2,
                                               const float* __restrict__ Kg,
                                               const float* __restrict__ x,
                                               const float* __restrict__ w_in,
                                               const float* __restrict__ b_in,
                                               const float* __restrict__ val_w,
                                               const float* __restrict__ val_b,
                                               const float* __restrict__ out_w,
                                               const float* __restrict__ out_b,
                                               const float* __restrict__ Afin,
                                               const float* __restrict__ Bfin,
                                               float* __restrict__ out) {
    __shared__ float sWv[4096], sWo[4096];
    __shared__ float sVb[64], sOb[64], sQw[64], sQb[64], sAf[64], sBf[64], sKg[64];
    __shared__ float sQV[8][64][16];          // 32 KB per-wave tiles
    int tid = threadIdx.x;
    size_t P0 = (size_t)blockIdx.x * 128;     // HW % 128 == 0 -> block in one batch
    int b = (int)(P0 / HW);
    for (int i = tid * 4; i < 4096; i += 256 * 4) {
        async_copy16(&sWv[i], &val_w[i]);
        async_copy16(&sWo[i], &out_w[i]);
    }
    if (tid < 64) {
        sVb[tid] = val_b[tid];
        sOb[tid] = out_b[tid];
        sQw[tid] = w_in[tid];
        sQb[tid] = b_in[tid];
        sAf[tid] = Afin[tid];
        sBf[tid] = Bfin[tid];
        sKg[tid] = Kg[b * 64 + tid] * (1.f / (float)HW);
    }
    async_wait0();
    __syncthreads();

    int wave = tid >> 5, lane = tid & 31, hi = lane >> 4, ln = lane & 15;
    size_t p0  = P0 + (size_t)wave * 16;
    int hw0    = (int)(p0 - (size_t)b * HW);
    size_t base = (size_t)b * 64 * HW + hw0 + ln;

    float xv = x[(size_t)b * HW + hw0 + ln];
    float g[4], mx = -1e30f;
#pragma unroll
    for (int i = 0; i < 4; ++i) {
        g[i] = xv * w_in[128 + i] + b_in[128 + i];
        mx = fmaxf(mx, g[i]);
    }
    float se = 0.f;
#pragma unroll
    for (int i = 0; i < 4; ++i) { g[i] = __expf(g[i] - mx); se += g[i]; }
    float gn0 = g[0] / se, gn1 = g[1] / se, gn2 = g[2] / se, gn3 = g[3] / se;

    // V = val_w @ K_final
    v8f accV[4];
#pragma unroll
    for (int m = 0; m < 4; ++m)
#pragma unroll
        for (int v = 0; v < 8; ++v) accV[m][v] = sVb[m * 16 + v + 8 * hi];
    for (int k0 = 0; k0 < 64; k0 += 4) {
        int c0 = k0 + 2 * hi, c1 = k0 + 1 + 2 * hi;
        size_t i0 = base + (size_t)c0 * HW, i1 = base + (size_t)c1 * HW;
        v2f bf;
        bf.x = gn0 * f0[i0] + gn1 * f1[i0] + gn2 * f2[i0] + gn3 * sKg[c0];
        bf.y = gn0 * f0[i1] + gn1 * f1[i1] + gn2 * f2[i1] + gn3 * sKg[c1];
#pragma unroll
        for (int m = 0; m < 4; ++m) {
            v2f af;
            af.x = sWv[(m * 16 + ln) * 64 + c0];
            af.y = sWv[(m * 16 + ln) * 64 + c1];
            accV[m] = __builtin_amdgcn_wmma_f32_16x16x4_f32(
                false, af, false, bf, (short)0, accV[m], false, false);
        }
    }
    // QV = Q * V -> LDS (D layout to B layout)
#pragma unroll
    for (int m = 0; m < 4; ++m)
#pragma unroll
        for (int v = 0; v < 8; ++v) {
            int ch = m * 16 + v + 8 * hi;
            float q = xv * sQw[ch] + sQb[ch];
            sQV[wave][ch][ln] = accV[m][v] * q;
        }
    __syncthreads();

    // O = out_w @ QV
    v8f accO[4];
#pragma unroll
    for (int m = 0; m < 4; ++m)
#pragma unroll
        for (int v = 0; v < 8; ++v) accO[m][v] = sOb[m * 16 + v + 8 * hi];
    for (int k0 = 0; k0 < 64; k0 += 4) {
        int c0 = k0 + 2 * hi, c1 = k0 + 1 + 2 * hi;
        v2f bf;
        bf.x = sQV[wave][c0][ln];
        bf.y = sQV[wave][c1][ln];
#pragma unroll
        for (int m = 0; m < 4; ++m) {
            v2f af;
            af.x = sWo[(m * 16 + ln) * 64 + c0];
            af.y = sWo[(m * 16 + ln) * 64 + c1];
            accO[m] = __builtin_amdgcn_wmma_f32_16x16x4_f32(
                false, af, false, bf, (short)0, accO[m], false, false);
        }
    }
    // out = O + folded x_proj
#pragma unroll
    for (int m = 0; m < 4; ++m)
#pragma unroll
        for (int v = 0; v < 8; ++v) {
            int ch = m * 16 + v + 8 * hi;
            out[base + (size_t)ch * HW] = accO[m][v] + sAf[ch] * xv + sBf[ch];
        }
}

// ---------------------------------------------------------------------------
// launcher
// ---------------------------------------------------------------------------
extern "C" void kernel_launch(void* const* d_in, const int* in_sizes, int n_in,
                              void* d_out, int out_size, void* d_ws, size_t ws_size,
                              hipStream_t stream) {
    const float* x      = (const float*)d_in[0];
    const float* w_in   = (const float*)d_in[1];
    const float* b_in   = (const float*)d_in[2];
    const float* dw[3]   = {(const float*)d_in[3],  (const float*)d_in[8],  (const float*)d_in[13]};
    const float* pw_w[3] = {(const float*)d_in[4],  (const float*)d_in[9],  (const float*)d_in[14]};
    const float* pw_b[3] = {(const float*)d_in[5],  (const float*)d_in[10], (const float*)d_in[15]};
    const float* bng[3]  = {(const float*)d_in[6],  (const float*)d_in[11], (const float*)d_in[16]};
    const float* bnb[3]  = {(const float*)d_in[7],  (const float*)d_in[12], (const float*)d_in[17]};
    const float* imp_w1 = (const float*)d_in[18];
    const float* imp_b1 = (const float*)d_in[19];
    const float* imp_w2 = (const float*)d_in[20];
    const float* imp_b2 = (const float*)d_in[21];
    const float* val_w  = (const float*)d_in[22];
    const float* val_b  = (const float*)d_in[23];
    const float* out_w  = (const float*)d_in[24];
    const float* out_b  = (const float*)d_in[25];
    const float* fin_w  = (const float*)d_in[26];
    const float* fin_b  = (const float*)d_in[27];
    float* outp = (float*)d_out;

    // workspace layout
    float* f    = (float*)d_ws;
    float* Kbuf = f;
    float* tmp  = f + (size_t)NTOT;
    float* ybuf = f + 2 * (size_t)NTOT;
    float* fA   = f + 3 * (size_t)NTOT;
    float* fB   = f + 4 * (size_t)NTOT;
    float* fC   = f + 5 * (size_t)NTOT;
    float* sm   = f + 6 * (size_t)NTOT;
    float* partS = sm;            sm += (size_t)TILEW * 64;
    float* partQ = sm;            sm += (size_t)TILEW * 64;
    float* imp   = sm;            sm += NPIX;
    float* mask  = sm;            sm += NPIX;
    float* alpha = sm;            sm += 16;
    float* beta  = sm;            sm += 16;
    float* Afin  = sm;            sm += 64;
    float* Bfin  = sm;            sm += 64;
    float* sumv  = sm;            sm += 64;
    float* sumsq = sm;            sm += 64;
    float* scale = sm;            sm += 64;
    float* shift = sm;            sm += 64;
    float* Kg    = sm;            sm += 512;
    unsigned* picks = (unsigned*)sm;            // 32 u32
    unsigned* hist  = (unsigned*)(sm + 32);     // 8*65536 u32

    const int PIXB  = NPIX / 256;               // 1568
    const int TOTB  = (int)(NTOT / 256);        // 100352
    const int TILEB = TILEW / 8;                // 3136 (8 waves/block)

    // 1) folded coefficients
    k_prep<<<1, 64, 0, stream>>>(w_in, b_in, imp_w1, imp_b1, fin_w, fin_b,
                                 alpha, beta, Afin, Bfin);
    // 2) importance map + exact deterministic top-k mask
    k_imp<<<PIXB, 256, 0, stream>>>(x, alpha, beta, imp_w2, imp_b2, imp);
    k_zero_u32<<<(8 * 65536 + 255) / 256, 256, 0, stream>>>(hist, 8 * 65536);
    k_hist_hi<<<PIXB, 256, 0, stream>>>(imp, hist);
    k_pick_hi<<<8, 32, 0, stream>>>(hist, picks);
    k_zero_u32<<<(8 * 65536 + 255) / 256, 256, 0, stream>>>(hist, 8 * 65536);
    k_hist_lo<<<PIXB, 256, 0, stream>>>(imp, picks, hist);
    k_pick_lo<<<8, 32, 0, stream>>>(hist, picks);
    k_mask<<<8, 1024, 0, stream>>>(imp, picks, mask);
    // 3) masked K planes
    k_writeKmasked<<<PIXB, 256, 0, stream>>>(x, mask, w_in, b_in, Kbuf);

    // 4) focal levels
    const int kszs[3] = {3, 5, 7};
    const int dils[3] = {1, 2, 4};
    float* fbufs[3] = {fA, fB, fC};
    const float* src = Kbuf;
    for (int l = 0; l < 3; ++l) {
        float* fcur = fbufs[l];
        k_dwconv<<<TOTB, 256, 0, stream>>>(src, tmp, dw[l], kszs[l], dils[l]);
        k_pw_wmma<<<TILEB, 256, 0, stream>>>(tmp, ybuf, pw_w[l], pw_b[l],
                                             partS, partQ);
        k_stats_red<<<64, 256, 0, stream>>>(partS, partQ, sumv, sumsq);
        k_bnparam<<<1, 64, 0, stream>>>(sumv, sumsq, bng[l], bnb[l], scale, shift);
        k_bnrelu<<<TOTB, 256, 0, stream>>>(ybuf, scale, shift, fcur);
        src = fcur;
    }
    // 5) per-(b,c) global means of focal2
    k_kgsum<<<512, 256, 0, stream>>>(fC, Kg);

    // 6) final fused WMMA kernel
    k_final<<<TILEB, 256, 0, stream>>>(fA, fB, fC, Kg, x, w_in, b_in,
                                       val_w, val_b, out_w, out_b,
                                       Afin, Bfin, outp);
}